// SumFusion_multiscale_62534723829937
// MI455X (gfx1250) — compile-verified
//
#include <hip/hip_runtime.h>
#include <hip/hip_bf16.h>
#include <stdint.h>

#define B_  4
#define L_  5
#define C_  64
#define H_  128
#define W_  256
#define HW_ (H_ * W_)
#define CB_ 16   // channel chunk (accumulators in VGPRs)

typedef unsigned int u32x4 __attribute__((ext_vector_type(4)));
typedef int          i32x8 __attribute__((ext_vector_type(8)));
typedef int          i32x4 __attribute__((ext_vector_type(4)));

__device__ __forceinline__ int clampi(int v, int lo, int hi) {
    return v < lo ? lo : (v > hi ? hi : v);
}

__global__ __launch_bounds__(W_) void warp_sum_kernel(
    const float* __restrict__ x,    // (B*L, C, H, W)
    const float* __restrict__ tm,   // (B, L, L, 2, 3)
    float* __restrict__ out)        // (B, C, H, W)
{
    const int w = threadIdx.x;      // 0..255
    const int h = blockIdx.x;       // 0..127
    const int b = blockIdx.y;       // 0..3

    __shared__ float s_theta[32];   // theta[b,0,l,i,j] : 30 floats

    // ---- Stage theta slab for this batch into LDS via the Tensor Data Mover ----
#if __has_builtin(__builtin_amdgcn_tensor_load_to_lds) && __has_builtin(__builtin_amdgcn_s_wait_tensorcnt)
    if (threadIdx.x < 32) {  // wave 0 issues the TDM op (EXEC-independent, per-wave)
        const uint64_t gaddr =
            (uint64_t)(uintptr_t)(tm + (size_t)b * (L_ * L_ * 6));
        const uint32_t lds_off = (uint32_t)(uintptr_t)(&s_theta[0]); // flat low 32 = LDS offset

        // D# group 0: [1:0]=count=1, [63:32]=lds_addr, [120:64]=global_addr, [127:126]=type=2
        const uint32_t d0 = (uint32_t)__builtin_amdgcn_readfirstlane((int)1u);
        const uint32_t d1 = (uint32_t)__builtin_amdgcn_readfirstlane((int)lds_off);
        const uint32_t d2 = (uint32_t)__builtin_amdgcn_readfirstlane((int)(uint32_t)gaddr);
        const uint32_t d3 = (uint32_t)__builtin_amdgcn_readfirstlane(
            (int)(((uint32_t)(gaddr >> 32) & 0x01FFFFFFu) | (2u << 30)));
        u32x4 g0 = { d0, d1, d2, d3 };

        // D# group 1: data_size=2 (4B) @bits[17:16]; tensor_dim0=30; tensor_dim1=1;
        // tile_dim0=30; tile_dim1=1; tensor_dim0_stride=30
        i32x8 g1 = { (int)0x00020000,
                     (int)(30u << 16),
                     (int)(1u  << 16),
                     (int)(30u << 16),
                     (int)1,
                     (int)30,
                     0, 0 };
        i32x4 z4 = { 0, 0, 0, 0 };
#if defined(__clang_major__) && (__clang_major__ >= 23)
        i32x8 z8 = { 0, 0, 0, 0, 0, 0, 0, 0 };
        __builtin_amdgcn_tensor_load_to_lds(g0, g1, z4, z4, z8, 0);
#else
        __builtin_amdgcn_tensor_load_to_lds(g0, g1, z4, z4, 0);
#endif
        __builtin_amdgcn_s_wait_tensorcnt(0);  // s_wait_tensorcnt 0
    }
#else
    if (threadIdx.x < 30)
        s_theta[threadIdx.x] = tm[(size_t)b * (L_ * L_ * 6) + threadIdx.x];
#endif
    __syncthreads();

    // ---- Per-(l,w) tap offsets + bilinear weights, reused across all 64 channels ----
    const float xs = (2.0f * (float)w + 1.0f) * (1.0f / (float)W_) - 1.0f;
    const float ys = (2.0f * (float)h + 1.0f) * (1.0f / (float)H_) - 1.0f;

    float wt[L_][4];
    int   off[L_][4];
#pragma unroll
    for (int l = 0; l < L_; ++l) {
        const float t0 = s_theta[l * 6 + 0], t1 = s_theta[l * 6 + 1], t2 = s_theta[l * 6 + 2];
        const float t3 = s_theta[l * 6 + 3], t4 = s_theta[l * 6 + 4], t5 = s_theta[l * 6 + 5];
        const float gx = t0 * xs + t1 * ys + t2;
        const float gy = t3 * xs + t4 * ys + t5;
        const float ix = ((gx + 1.0f) * (float)W_ - 1.0f) * 0.5f;
        const float iy = ((gy + 1.0f) * (float)H_ - 1.0f) * 0.5f;
        const float x0f = floorf(ix), y0f = floorf(iy);
        const float wx1 = ix - x0f, wx0 = 1.0f - wx1;
        const float wy1 = iy - y0f, wy0 = 1.0f - wy1;
        const int x0 = (int)x0f, y0 = (int)y0f;
        const int x1 = x0 + 1,   y1 = y0 + 1;
        const bool vx0 = (x0 >= 0) && (x0 < W_);
        const bool vx1 = (x1 >= 0) && (x1 < W_);
        const bool vy0 = (y0 >= 0) && (y0 < H_);
        const bool vy1 = (y1 >= 0) && (y1 < H_);
        const int x0c = clampi(x0, 0, W_ - 1), x1c = clampi(x1, 0, W_ - 1);
        const int y0c = clampi(y0, 0, H_ - 1), y1c = clampi(y1, 0, H_ - 1);
        wt[l][0] = (vy0 && vx0) ? wy0 * wx0 : 0.0f;  off[l][0] = y0c * W_ + x0c;
        wt[l][1] = (vy0 && vx1) ? wy0 * wx1 : 0.0f;  off[l][1] = y0c * W_ + x1c;
        wt[l][2] = (vy1 && vx0) ? wy1 * wx0 : 0.0f;  off[l][2] = y1c * W_ + x0c;
        wt[l][3] = (vy1 && vx1) ? wy1 * wx1 : 0.0f;  off[l][3] = y1c * W_ + x1c;
    }

    // ---- Gather + accumulate: 4 chunks of 16 channels ----
    // Two-phase (load-all, then fma-all) per (l, chunk): lets the scheduler
    // clause 64 global_load_b32 sharing 4 base addresses + 24-bit immediate
    // offsets (i*HW_*4 <= 1.97MB), and hide latency behind one wait.
    const float* xb = x + (size_t)(b * L_) * C_ * HW_;

    for (int c0 = 0; c0 < C_; c0 += CB_) {
        float acc[CB_];
#pragma unroll
        for (int i = 0; i < CB_; ++i) acc[i] = 0.0f;

#pragma unroll
        for (int l = 0; l < L_; ++l) {
            const float* p  = xb + ((size_t)l * C_ + c0) * HW_;
            const float* b0 = p + off[l][0];
            const float* b1 = p + off[l][1];
            const float* b2 = p + off[l][2];
            const float* b3 = p + off[l][3];

            float v0[CB_], v1[CB_], v2[CB_], v3[CB_];
#pragma unroll
            for (int i = 0; i < CB_; ++i) {
                v0[i] = b0[(size_t)i * HW_];
                v1[i] = b1[(size_t)i * HW_];
                v2[i] = b2[(size_t)i * HW_];
                v3[i] = b3[(size_t)i * HW_];
            }

            const float w0 = wt[l][0], w1 = wt[l][1], w2 = wt[l][2], w3 = wt[l][3];
#pragma unroll
            for (int i = 0; i < CB_; ++i) {
                float a = acc[i];
                a = fmaf(w0, v0[i], a);
                a = fmaf(w1, v1[i], a);
                a = fmaf(w2, v2[i], a);
                a = fmaf(w3, v3[i], a);
                acc[i] = a;
            }
        }

#pragma unroll
        for (int i = 0; i < CB_; ++i) {
            // streaming output: non-temporal so x stays resident in the 192MB L2
            __builtin_nontemporal_store(
                acc[i],
                out + (((size_t)b * C_ + (size_t)(c0 + i)) * H_ + h) * W_ + w);
        }
    }
}

extern "C" void kernel_launch(void* const* d_in, const int* in_sizes, int n_in,
                              void* d_out, int out_size, void* d_ws, size_t ws_size,
                              hipStream_t stream) {
    const float* x  = (const float*)d_in[0];
    // d_in[1] = record_len (int64), unused by the reference computation
    const float* tm = (const float*)d_in[2];
    float* out = (float*)d_out;

    dim3 grid(H_, B_);   // (128, 4) workgroups
    dim3 block(W_);      // 256 threads = 8 wave32
    warp_sum_kernel<<<grid, block, 0, stream>>>(x, tm, out);
}